// hungarian_matcher_24369644438189
// MI455X (gfx1250) — compile-verified
//
#include <hip/hip_runtime.h>

// DETR Hungarian-matcher cost on gfx1250 (MI455X, wave32).
//
// cost[b,q,t] = 5 * sum_d |pbox[b,q,d] - tbox[b*64+t,d]|  (d = 0..10)
//             - softmax(logits[b,q,:])[cls[b*64+t]]
//
// Class term is a K=4 GEMM against a one-hot matrix, fused into the bbox term
// through the WMMA C accumulator:
//   D(16x16) = A(-prob, 16x4) x B(onehot, 4x16) + C(5*L1, 16x16)
// -> one v_wmma_f32_16x16x4_f32 per 16-query x 16-target output tile.
// COST_BBOX (=5) is folded into the LDS staging (|5q - 5t| = 5|q - t|), so
// the accumulator holds the final bbox term with no epilogue scaling.

typedef __attribute__((ext_vector_type(2))) float v2f;
typedef __attribute__((ext_vector_type(8))) float v8f;

#define BS   16
#define NQ   900
#define NC   4
#define BD   11
#define PER  64
#define QT   16      // queries per block (one WMMA M-tile)
#define NTHR 128     // 4 waves; wave w handles target tile t0 = 16*w
#define WBBOX 5.0f

__global__ __launch_bounds__(NTHR) void matcher_cost_kernel(
    const float* __restrict__ logits,   // [16,900,4]
    const float* __restrict__ pboxes,   // [16,900,11]
    const float* __restrict__ tboxes,   // [1024,11]
    const int*   __restrict__ tids,     // [1024]
    float*       __restrict__ out)      // [16,900,64]
{
    __shared__ float s_tbox[BD][PER];     // transposed, premultiplied by 5
    __shared__ float s_qbox[QT][BD + 1];  // premultiplied by 5, padded stride
    __shared__ float s_prob[QT][NC];      // softmax probs per query row
    __shared__ int   s_cls[PER];          // target class ids for this batch

    const int b    = blockIdx.x;
    const int q0   = blockIdx.y * QT;
    const int tid  = threadIdx.x;
    const int lane = tid & 31;
    const int wave = tid >> 5;

    // ---- stage batch-b targets: boxes (transposed, x5) + class ids ---------
    for (int i = tid; i < BD * PER; i += NTHR) {
        const int d = i >> 6, t = i & (PER - 1);
        s_tbox[d][t] = WBBOX * tboxes[(size_t)(b * PER + t) * BD + d];
    }
    if (tid < PER) s_cls[tid] = tids[b * PER + tid];

    // ---- stage 16 query boxes (x5, clamped at the q=900 edge) --------------
    for (int i = tid; i < QT * BD; i += NTHR) {
        const int q = i / BD, d = i % BD;
        int qq = q0 + q; if (qq > NQ - 1) qq = NQ - 1;
        s_qbox[q][d] = WBBOX * pboxes[((size_t)b * NQ + qq) * BD + d];
    }

    // ---- softmax over 4 classes, one query per lane (16 lanes) -------------
    if (tid < QT) {
        int qq = q0 + tid; if (qq > NQ - 1) qq = NQ - 1;
        const float* lg = logits + ((size_t)b * NQ + qq) * NC;
        const float l0 = lg[0], l1 = lg[1], l2 = lg[2], l3 = lg[3];
        const float m  = fmaxf(fmaxf(l0, l1), fmaxf(l2, l3));
        const float e0 = __expf(l0 - m), e1 = __expf(l1 - m);
        const float e2 = __expf(l2 - m), e3 = __expf(l3 - m);
        const float inv = 1.0f / (e0 + e1 + e2 + e3);
        s_prob[tid][0] = e0 * inv;  s_prob[tid][1] = e1 * inv;
        s_prob[tid][2] = e2 * inv;  s_prob[tid][3] = e3 * inv;
    }
    __syncthreads();

    const int t0 = wave * 16;     // this wave's 16-target N-tile
    const int n  = lane & 15;
    const int hi = lane >> 4;     // lane half selects K-pair / +8 row offset

    // A (16x4 f32, 2 VGPRs): lanes 0-15 hold K=0,1; lanes 16-31 hold K=2,3.
    // Row M = lane&15. Class weight COST_CLASS=1, negated gather => A = -prob.
    v2f A;
    A.x = -s_prob[n][hi * 2 + 0];
    A.y = -s_prob[n][hi * 2 + 1];

    // B (4x16 one-hot, 2 VGPRs): column N = lane&15, same K-pair split as A.
    const int cls = s_cls[t0 + n];
    v2f B;
    B.x = (cls == hi * 2 + 0) ? 1.0f : 0.0f;
    B.y = (cls == hi * 2 + 1) ? 1.0f : 0.0f;

    // C (16x16 f32, 8 VGPRs): VGPR r -> row M = r + 8*hi, col N = lane&15.
    // Boxes are pre-scaled by 5, so this accumulates the final bbox term.
    v8f C = {};
    for (int d = 0; d < BD; ++d) {
        const float tb = s_tbox[d][t0 + n];
        #pragma unroll
        for (int r = 0; r < 8; ++r)
            C[r] += fabsf(s_qbox[r + hi * 8][d] - tb);
    }

    // D = A*B + C : fused class gather + weighted bbox cost, one matrix op.
    // EXEC is all-ones here (tail masking happens only after the WMMA).
    v8f D = __builtin_amdgcn_wmma_f32_16x16x4_f32(
        /*neg_a=*/false, A, /*neg_b=*/false, B,
        /*c_mod=*/(short)0, C, /*reuse_a=*/false, /*reuse_b=*/false);

    // Store: one base address, rows differ by PER floats (256 B immediate
    // offsets). Bounds check is block-uniform -> scalar branch, no exec mask
    // churn except in the single partial tail tile (blockIdx.y == 56).
    float* base = out + (((size_t)b * NQ + q0 + hi * 8) * PER + t0 + n);
    if (q0 + QT <= NQ) {
        #pragma unroll
        for (int r = 0; r < 8; ++r)
            base[(size_t)r * PER] = D[r];
    } else {
        #pragma unroll
        for (int r = 0; r < 8; ++r)
            if (q0 + hi * 8 + r < NQ)
                base[(size_t)r * PER] = D[r];
    }
}

extern "C" void kernel_launch(void* const* d_in, const int* in_sizes, int n_in,
                              void* d_out, int out_size, void* d_ws, size_t ws_size,
                              hipStream_t stream) {
    const float* logits = (const float*)d_in[0];  // pred_logits  [16,900,4]
    const float* pboxes = (const float*)d_in[1];  // pred_boxes   [16,900,11]
    const float* tboxes = (const float*)d_in[2];  // target_boxes [1024,11]
    const int*   tids   = (const int*)d_in[3];    // target_ids   [1024]
    // d_in[4] = size_per_batch scalar (== 64, baked in as PER)
    (void)in_sizes; (void)n_in; (void)out_size; (void)d_ws; (void)ws_size;

    dim3 grid(BS, (NQ + QT - 1) / QT);   // 16 x 57 blocks
    matcher_cost_kernel<<<grid, NTHR, 0, stream>>>(
        logits, pboxes, tboxes, tids, (float*)d_out);
}